// GeeseNetA_69337952026708
// MI455X (gfx1250) — compile-verified
//
#include <hip/hip_runtime.h>
#include <cstddef>
#include <cstdint>

// ---------- CDNA5 WMMA: full-precision f32 16x16x4 ----------
typedef float v2f __attribute__((ext_vector_type(2)));
typedef float v8f __attribute__((ext_vector_type(8)));

__device__ __forceinline__ v8f wmma_f32(v2f a, v2f b, v8f c) {
  // 8-arg form: (neg_a, A, neg_b, B, c_mod, C, reuse_a, reuse_b)
  return __builtin_amdgcn_wmma_f32_16x16x4_f32(false, a, false, b, (short)0, c, false, false);
}

__device__ __forceinline__ float sigf(float x) { return 1.0f / (1.0f + __expf(-x)); }

#define S77 77
#define SP 80           // padded S
#define BB 1024
#define DD 32
#define MROWS (S77 * BB)   // 78848, multiple of 16
#define GI_CHUNKS (S77 * 24)  // 77 rows * 24 x 16B chunks = 1848

// ---------- 1) encoder: 1x1 conv + BN(inference) + positional encoding ----------
__global__ void encoder_kernel(const float* __restrict__ x, const float* __restrict__ cw,
                               const float* __restrict__ cb, const float* __restrict__ bg,
                               const float* __restrict__ bb, const float* __restrict__ bm,
                               const float* __restrict__ bv, float* __restrict__ src) {
  int b = blockIdx.x;                 // 0..1023
  int d = threadIdx.x & 31;
  int s0 = threadIdx.x >> 5;          // 0..7
  float scale = bg[d] * rsqrtf(bv[d] + 1e-5f);
  float freq = __expf(-(float)(d & ~1) * (9.210340371976184f / 32.0f)); // ln(1e4)/D
  for (int s = s0; s < S77; s += 8) {
    float acc = cb[d];
    #pragma unroll
    for (int c = 0; c < 17; ++c)
      acc += x[((size_t)b * 17 + c) * S77 + s] * cw[d * 17 + c];
    acc = (acc - bm[d]) * scale + bb[d];
    float ang = (float)s * freq;
    acc += (d & 1) ? cosf(ang) : sinf(ang);
    src[((size_t)s * BB + b) * DD + d] = acc;
  }
}

// ---------- 2) h0[s,d] = sum_b src[s,b,d] ----------
__global__ void h0_reduce_kernel(const float* __restrict__ src, float* __restrict__ h0) {
  __shared__ float red[8][32];
  int s = blockIdx.x, d = threadIdx.x & 31, c = threadIdx.x >> 5;
  float acc = 0.f;
  for (int b = c; b < BB; b += 8) acc += src[((size_t)s * BB + b) * DD + d];
  red[c][d] = acc;
  __syncthreads();
  if (c == 0) {
    float t = 0.f;
    #pragma unroll
    for (int j = 0; j < 8; ++j) t += red[j][d];
    h0[s * DD + d] = t;
  }
}

// ---------- 3) in-place LayerNorm over D=32, one wave per row ----------
__global__ void layernorm_kernel(float* __restrict__ X, const float* __restrict__ g,
                                 const float* __restrict__ bta) {
  size_t row = blockIdx.x;
  int lane = threadIdx.x;
  float xv = X[row * DD + lane];
  float m = xv;
  for (int o = 16; o; o >>= 1) m += __shfl_xor(m, o, 32);
  m *= (1.0f / 32.0f);
  float dx = xv - m;
  float v = dx * dx;
  for (int o = 16; o; o >>= 1) v += __shfl_xor(v, o, 32);
  v *= (1.0f / 32.0f);
  X[row * DD + lane] = dx * rsqrtf(v + 1e-5f) * g[lane] + bta[lane];
}

// ---------- 4) generic GEMM: C[M,N] = act(A[M,K] @ W[N,K]^T + bias) ----------
// one wave per 16x16 output tile; M % 16 == 0, N % 16 == 0, K % 4 == 0
template <int ACT>
__global__ void gemm_wt_kernel(const float* __restrict__ A, const float* __restrict__ W,
                               const float* __restrict__ bias, float* __restrict__ C,
                               int N, int K) {
  int lane = threadIdx.x;
  int hi = lane >> 4, l = lane & 15;
  int n0 = blockIdx.x * 16;
  size_t m0 = (size_t)blockIdx.y * 16;
  const float* Ar = A + (m0 + l) * K;
  const float* Wr = W + (size_t)(n0 + l) * K;
  v8f acc = {};
  for (int k = 0; k < K; k += 4) {
    v2f a, b;
    a.x = Ar[k + 2 * hi]; a.y = Ar[k + 2 * hi + 1];
    b.x = Wr[k + 2 * hi]; b.y = Wr[k + 2 * hi + 1];
    acc = wmma_f32(a, b, acc);
  }
  float bn = bias ? bias[n0 + l] : 0.0f;
  #pragma unroll
  for (int j = 0; j < 8; ++j) {
    float v = acc[j] + bn;
    if (ACT) v = fmaxf(v, 0.0f);
    C[(m0 + j + 8 * hi) * N + n0 + l] = v;
  }
}

// ---------- 5) fused attention per (batch, head): 1 wave ----------
__global__ void attention_kernel(const float* __restrict__ qkv, float* __restrict__ o) {
  __shared__ float Q[SP][8], Kl[SP][8], Vl[SP][16], Sc[SP][SP];
  int b = blockIdx.x >> 2, hh = blockIdx.x & 3;
  int lane = threadIdx.x;
  int hi = lane >> 4, l = lane & 15;
  for (int s = lane; s < SP; s += 32) {
    const float* base = qkv + ((size_t)s * BB + b) * 96 + hh * 8;
    bool ok = s < S77;
    #pragma unroll
    for (int d = 0; d < 8; ++d) {
      Q[s][d]  = ok ? base[d] * 0.3535533905932738f : 0.f;  // 1/sqrt(8)
      Kl[s][d] = ok ? base[32 + d] : 0.f;
      Vl[s][d] = ok ? base[64 + d] : 0.f;
      Vl[s][8 + d] = 0.f;
    }
  }
  __syncthreads();
  // scores (80x80) = Q @ K^T, K-dim = 8
  for (int mt = 0; mt < 5; ++mt)
    for (int nt = 0; nt < 5; ++nt) {
      v8f acc = {};
      #pragma unroll
      for (int kk = 0; kk < 8; kk += 4) {
        v2f a, bfr;
        a.x = Q[mt * 16 + l][kk + 2 * hi];   a.y = Q[mt * 16 + l][kk + 2 * hi + 1];
        bfr.x = Kl[nt * 16 + l][kk + 2 * hi]; bfr.y = Kl[nt * 16 + l][kk + 2 * hi + 1];
        acc = wmma_f32(a, bfr, acc);
      }
      #pragma unroll
      for (int j = 0; j < 8; ++j) {
        int row = mt * 16 + j + 8 * hi, col = nt * 16 + l;
        Sc[row][col] = (col < S77) ? acc[j] : -1e30f;
      }
    }
  __syncthreads();
  // softmax over valid cols
  for (int s = lane; s < SP; s += 32) {
    float mx = -1e30f;
    for (int t = 0; t < S77; ++t) mx = fmaxf(mx, Sc[s][t]);
    float sum = 0.f;
    for (int t = 0; t < S77; ++t) { float e = __expf(Sc[s][t] - mx); Sc[s][t] = e; sum += e; }
    float inv = 1.0f / sum;
    for (int t = 0; t < S77; ++t) Sc[s][t] *= inv;
    for (int t = S77; t < SP; ++t) Sc[s][t] = 0.f;
  }
  __syncthreads();
  // O = attn @ V, K-dim = 80, one 16-wide N tile (cols 8..15 are zero pad)
  for (int mt = 0; mt < 5; ++mt) {
    v8f acc = {};
    for (int kk = 0; kk < SP; kk += 4) {
      v2f a, bfr;
      a.x = Sc[mt * 16 + l][kk + 2 * hi];  a.y = Sc[mt * 16 + l][kk + 2 * hi + 1];
      bfr.x = Vl[kk + 2 * hi][l];          bfr.y = Vl[kk + 2 * hi + 1][l];
      acc = wmma_f32(a, bfr, acc);
    }
    #pragma unroll
    for (int j = 0; j < 8; ++j) {
      int s = mt * 16 + j + 8 * hi;
      if (s < S77 && l < 8) o[((size_t)s * BB + b) * DD + hh * 8 + l] = acc[j];
    }
  }
}

// ---------- 6) persistent GRU scan: 1 workgroup, 8 waves ----------
// gi = x_t@Wih^T + bih precomputed in global. Per step:
//   - async-copy next step's gi slice (77x96 f32) global->LDS (double buffered)
//   - gh = h@Whh^T via f32 WMMA from LDS-resident h and Whh
//   - gates from LDS-only operands; h ping-pongs between two LDS buffers
__global__ void gru_kernel(const float* __restrict__ gi, const float* __restrict__ whh,
                           const float* __restrict__ bhh, const float* __restrict__ h0,
                           float* __restrict__ out, float* __restrict__ h_last) {
  __shared__ float GiL[2][S77 * 96];   // 59136 B : double-buffered input gates
  __shared__ float Gh[SP][96];         // 30720 B : gh scratch
  __shared__ float Hs[2][SP][32];      // 20480 B : ping-pong hidden state
  __shared__ float Wh[96 * 32];        // 12288 B : Whh staged once
  __shared__ float Bh[96];             //   384 B : bhh staged once
  int tid = threadIdx.x;
  int wave = tid >> 5, lane = tid & 31;
  int hi = lane >> 4, l = lane & 15;

  // kick off async copy of step-0 gi while we stage weights/state
  for (int c = tid; c < GI_CHUNKS; c += 256) {
    int s = c / 24, off = (c % 24) * 4;
    uint64_t ga = (uint64_t)(gi + ((size_t)s * BB) * 96 + off);
    uint32_t la = (uint32_t)(uintptr_t)(&GiL[0][s * 96 + off]);
    asm volatile("global_load_async_to_lds_b128 %0, %1, off" :: "v"(la), "v"(ga) : "memory");
  }
  for (int i = tid; i < 96 * 32; i += 256) Wh[i] = whh[i];
  if (tid < 96) Bh[tid] = bhh[tid];
  for (int i = tid; i < SP * 32; i += 256) {
    int s = i >> 5, d = i & 31;
    float v = (s < S77) ? h0[s * 32 + d] : 0.f;
    Hs[0][s][d] = v;
    Hs[1][s][d] = 0.f;  // pad rows stay 0; real rows overwritten by gates
  }
  asm volatile("s_wait_asynccnt 0" ::: "memory");
  __syncthreads();

  int p = 0;
  for (int b = 0; b < BB; ++b) {
    // prefetch gi for step b+1 into the other buffer (overlaps with compute)
    if (b + 1 < BB) {
      const float* gnext = gi + (size_t)(b + 1) * 96;
      float* dst = &GiL[(b + 1) & 1][0];
      for (int c = tid; c < GI_CHUNKS; c += 256) {
        int s = c / 24, off = (c % 24) * 4;
        uint64_t ga = (uint64_t)(gnext + ((size_t)s * BB) * 96 + off);
        uint32_t la = (uint32_t)(uintptr_t)(dst + s * 96 + off);
        asm volatile("global_load_async_to_lds_b128 %0, %1, off" :: "v"(la), "v"(ga) : "memory");
      }
    }
    // gh tiles: 5 (M) x 6 (N), K=32 -> 30 tiles over 8 waves, all-LDS operands
    for (int t = wave; t < 30; t += 8) {
      int mt = t / 6, nt = t % 6;
      v8f acc = {};
      #pragma unroll
      for (int kk = 0; kk < 32; kk += 4) {
        v2f a, bfr;
        a.x = Hs[p][mt * 16 + l][kk + 2 * hi];
        a.y = Hs[p][mt * 16 + l][kk + 2 * hi + 1];
        bfr.x = Wh[(nt * 16 + l) * 32 + kk + 2 * hi];
        bfr.y = Wh[(nt * 16 + l) * 32 + kk + 2 * hi + 1];
        acc = wmma_f32(a, bfr, acc);
      }
      #pragma unroll
      for (int j = 0; j < 8; ++j)
        Gh[mt * 16 + j + 8 * hi][nt * 16 + l] = acc[j];
    }
    __syncthreads();
    // gates: LDS-only reads; write new h into the other ping-pong buffer
    const float* gib = &GiL[b & 1][0];
    for (int i = tid; i < S77 * 32; i += 256) {
      int s = i >> 5, d = i & 31;
      const float* gir = gib + s * 96;
      float r = sigf(gir[d]      + Gh[s][d]      + Bh[d]);
      float z = sigf(gir[32 + d] + Gh[s][32 + d] + Bh[32 + d]);
      float n = tanhf(gir[64 + d] + r * (Gh[s][64 + d] + Bh[64 + d]));
      float hn = (1.0f - z) * n + z * Hs[p][s][d];
      out[((size_t)s * BB + b) * 32 + d] = hn;
      Hs[p ^ 1][s][d] = hn;
    }
    // retire this wave's async loads for step b+1, then make them visible
    asm volatile("s_wait_asynccnt 0" ::: "memory");
    __syncthreads();
    p ^= 1;
  }
  if (h_last)
    for (int i = tid; i < S77 * 32; i += 256) {
      int s = i >> 5, d = i & 31;
      h_last[s * 32 + d] = Hs[p][s][d];
    }
}

// ---------- 7) heads ----------
__global__ void heads_kernel(const float* __restrict__ out2, const float* __restrict__ hp,
                             const float* __restrict__ hv, float* __restrict__ dout) {
  int b = blockIdx.x * blockDim.x + threadIdx.x;
  if (b >= BB) return;
  const float* row38 = out2 + ((size_t)38 * BB + b) * 32;
  #pragma unroll
  for (int i = 0; i < 4; ++i) {
    float acc = 0.f;
    #pragma unroll
    for (int d = 0; d < 32; ++d) acc += row38[d] * hp[i * 32 + d];
    dout[b * 4 + i] = acc;
  }
  float vacc = 0.f;
  for (int s = 0; s < S77; ++s) {
    const float* r = out2 + ((size_t)s * BB + b) * 32;
    float m = 0.f;
    #pragma unroll
    for (int d = 0; d < 32; ++d) m += r[d];
    vacc += (m * (1.0f / 32.0f)) * hv[s];
  }
  dout[4 * BB + b] = tanhf(vacc);
}

// ---------- launch ----------
extern "C" void kernel_launch(void* const* d_in, const int* in_sizes, int n_in,
                              void* d_out, int out_size, void* d_ws, size_t ws_size,
                              hipStream_t stream) {
  const float* x         = (const float*)d_in[0];
  const float* conv_w    = (const float*)d_in[1];
  const float* conv_b    = (const float*)d_in[2];
  const float* bn_g      = (const float*)d_in[3];
  const float* bn_b      = (const float*)d_in[4];
  const float* bn_m      = (const float*)d_in[5];
  const float* bn_v      = (const float*)d_in[6];
  const float* in_proj_w = (const float*)d_in[7];
  const float* in_proj_b = (const float*)d_in[8];
  const float* out_proj_w= (const float*)d_in[9];
  const float* out_proj_b= (const float*)d_in[10];
  const float* ln1_g     = (const float*)d_in[11];
  const float* ln1_b     = (const float*)d_in[12];
  const float* ln2_g     = (const float*)d_in[13];
  const float* ln2_b     = (const float*)d_in[14];
  const float* g1_wih    = (const float*)d_in[15];
  const float* g1_whh    = (const float*)d_in[16];
  const float* g1_bih    = (const float*)d_in[17];
  const float* g1_bhh    = (const float*)d_in[18];
  const float* g2_wih    = (const float*)d_in[19];
  const float* g2_whh    = (const float*)d_in[20];
  const float* g2_bih    = (const float*)d_in[21];
  const float* g2_bhh    = (const float*)d_in[22];
  const float* lin1_w    = (const float*)d_in[23];
  const float* lin1_b    = (const float*)d_in[24];
  const float* lin2_w    = (const float*)d_in[25];
  const float* lin2_b    = (const float*)d_in[26];
  const float* head_p_w  = (const float*)d_in[27];
  const float* head_v_w  = (const float*)d_in[28];

  float* ws = (float*)d_ws;
  const size_t SZ32 = (size_t)MROWS * 32;    // 2,523,136
  float* srcb  = ws;                          // src -> gru1_out (ln2'd in place)
  float* big   = ws + SZ32;                   // qkv/gi1 -> ffn1(78848x128) -> gi2
  float* obuf  = big + (size_t)MROWS * 128;   // attn o -> ffn2
  float* o2buf = obuf + SZ32;                 // out_proj -> gru2_out
  float* h0b   = o2buf + SZ32;                // 77*32
  float* hlast = h0b + S77 * 32;              // 77*32

  const int MT = MROWS / 16;                  // 4928 tiles in M

  encoder_kernel<<<BB, 256, 0, stream>>>(x, conv_w, conv_b, bn_g, bn_b, bn_m, bn_v, srcb);
  h0_reduce_kernel<<<S77, 256, 0, stream>>>(srcb, h0b);
  layernorm_kernel<<<MROWS, 32, 0, stream>>>(srcb, ln1_g, ln1_b);
  gemm_wt_kernel<0><<<dim3(6, MT), 32, 0, stream>>>(srcb, in_proj_w, in_proj_b, big, 96, 32);
  attention_kernel<<<BB * 4, 32, 0, stream>>>(big, obuf);
  gemm_wt_kernel<0><<<dim3(2, MT), 32, 0, stream>>>(obuf, out_proj_w, out_proj_b, o2buf, 32, 32);
  gemm_wt_kernel<0><<<dim3(6, MT), 32, 0, stream>>>(o2buf, g1_wih, g1_bih, big, 96, 32);
  gru_kernel<<<1, 256, 0, stream>>>(big, g1_whh, g1_bhh, h0b, srcb, hlast);
  layernorm_kernel<<<MROWS, 32, 0, stream>>>(srcb, ln2_g, ln2_b);
  gemm_wt_kernel<1><<<dim3(8, MT), 32, 0, stream>>>(srcb, lin1_w, lin1_b, big, 128, 32);
  gemm_wt_kernel<1><<<dim3(2, MT), 32, 0, stream>>>(big, lin2_w, lin2_b, obuf, 32, 128);
  gemm_wt_kernel<0><<<dim3(6, MT), 32, 0, stream>>>(obuf, g2_wih, g2_bih, big, 96, 32);
  gru_kernel<<<1, 256, 0, stream>>>(big, g2_whh, g2_bhh, hlast, o2buf, nullptr);
  heads_kernel<<<4, 256, 0, stream>>>(o2buf, head_p_w, head_v_w, (float*)d_out);
}